// mincutnet_71846212927560
// MI455X (gfx1250) — compile-verified
//
#include <hip/hip_runtime.h>
#include <hip/hip_bf16.h>

typedef float v2f __attribute__((ext_vector_type(2)));
typedef float v8f __attribute__((ext_vector_type(8)));

#define BB   32
#define NN   2048
#define BN   (BB * NN)      // 65536 nodes
#define INC  128
#define HIDC 16
#define KC   20
#define OUTC 32
#define EPB  2048           // edges per block (out_adj kernel)
#define ECH  64             // edge staging chunk

// ---------------------------------------------------------------- utilities
__global__ void fill_zero_kernel(float* __restrict__ p, int n) {
    int i = blockIdx.x * blockDim.x + threadIdx.x;
    if (i < n) p[i] = 0.0f;
}

// ---------------------------------------------------- GEMM1: h0 = x @ W1 (WMMA f32)
// grid 512 x 256 threads (8 waves); each wave = one 16-row M tile, N=16, K=128.
__global__ void gemm1_wmma_kernel(const float* __restrict__ x,
                                  const float* __restrict__ W1,
                                  float* __restrict__ h0) {
    const int lane = threadIdx.x & 31;
    const int wave = threadIdx.x >> 5;
    const int tile = blockIdx.x * 8 + wave;
    const int m    = lane & 15;
    const bool hi  = lane >= 16;
    const int row  = tile * 16 + m;
    const float* arow = x + row * INC + (hi ? 2 : 0);

    v8f c = {0.f, 0.f, 0.f, 0.f, 0.f, 0.f, 0.f, 0.f};
#pragma unroll
    for (int k0 = 0; k0 < INC; k0 += 4) {
        v2f a; a.x = arow[k0]; a.y = arow[k0 + 1];
        const int kb = k0 + (hi ? 2 : 0);
        v2f b; b.x = W1[kb * HIDC + m]; b.y = W1[(kb + 1) * HIDC + m];
        c = __builtin_amdgcn_wmma_f32_16x16x4_f32(false, a, false, b,
                                                  (short)0, c, false, false);
    }
    const int mbase = hi ? 8 : 0;
    float* orow = h0 + (tile * 16) * HIDC + m;
#pragma unroll
    for (int r = 0; r < 8; ++r) orow[(mbase + r) * HIDC] = c[r];
}

// ------------------------------------------------------------ degree histogram
__global__ void deg_kernel(const int* __restrict__ dst, float* __restrict__ deg, int E) {
    int e = blockIdx.x * blockDim.x + threadIdx.x;
    if (e < E) atomicAdd(&deg[dst[e]], 1.0f);
}

// dinv = rsqrt(deg + 1 self-loop); init agg with self-loop contribution
__global__ void dinv_self_kernel(const float* __restrict__ deg,
                                 float* __restrict__ dinv,
                                 const float* __restrict__ h0,
                                 float* __restrict__ agg) {
    int i = blockIdx.x * blockDim.x + threadIdx.x;
    if (i >= BN) return;
    float di = rsqrtf(deg[i] + 1.0f);
    dinv[i] = di;
    float w = di * di;
#pragma unroll
    for (int c = 0; c < HIDC; ++c) agg[i * HIDC + c] = h0[i * HIDC + c] * w;
}

// scatter-add: agg[dst] += h0[src] * dinv[src]*dinv[dst]   (thread per (edge,ch))
__global__ void scatter_kernel(const int* __restrict__ src, const int* __restrict__ dst,
                               const float* __restrict__ dinv, const float* __restrict__ h0,
                               float* __restrict__ agg, int E) {
    int tid = blockIdx.x * blockDim.x + threadIdx.x;
    if (tid >= E * HIDC) return;
    int e = tid >> 4, c = tid & 15;
    int si = src[e], di = dst[e];
    float nrm = dinv[si] * dinv[di];
    atomicAdd(&agg[di * HIDC + c], h0[si * HIDC + c] * nrm);
}

__global__ void bias_relu_kernel(const float* __restrict__ agg,
                                 const float* __restrict__ b1,
                                 float* __restrict__ nemb) {
    int tid = blockIdx.x * blockDim.x + threadIdx.x;
    if (tid >= BN * HIDC) return;
    nemb[tid] = fmaxf(agg[tid] + b1[tid & 15], 0.0f);
}

// ----------------------------------- s = node_emb @ Wp + bp (WMMA f32, N padded to 32)
__global__ void sgemm_wmma_kernel(const float* __restrict__ ne,
                                  const float* __restrict__ Wp,
                                  const float* __restrict__ bp,
                                  float* __restrict__ s) {
    const int lane = threadIdx.x & 31;
    const int wave = threadIdx.x >> 5;
    const int tile = blockIdx.x * 8 + wave;
    const int m    = lane & 15;
    const bool hi  = lane >= 16;
    const float* arow = ne + (tile * 16 + m) * HIDC + (hi ? 2 : 0);
    const bool c1v = (16 + m) < KC;   // cols 16..19 only

    v8f c0 = {0.f, 0.f, 0.f, 0.f, 0.f, 0.f, 0.f, 0.f};
    v8f c1 = {0.f, 0.f, 0.f, 0.f, 0.f, 0.f, 0.f, 0.f};
#pragma unroll
    for (int k0 = 0; k0 < HIDC; k0 += 4) {
        v2f a; a.x = arow[k0]; a.y = arow[k0 + 1];
        const int kb = k0 + (hi ? 2 : 0);
        v2f b0; b0.x = Wp[kb * KC + m]; b0.y = Wp[(kb + 1) * KC + m];
        v2f b1; b1.x = c1v ? Wp[kb * KC + 16 + m] : 0.f;
                b1.y = c1v ? Wp[(kb + 1) * KC + 16 + m] : 0.f;
        c0 = __builtin_amdgcn_wmma_f32_16x16x4_f32(false, a, false, b0, (short)0, c0, false, false);
        c1 = __builtin_amdgcn_wmma_f32_16x16x4_f32(false, a, false, b1, (short)0, c1, false, false);
    }
    const int mbase = hi ? 8 : 0;
#pragma unroll
    for (int r = 0; r < 8; ++r) {
        int row = tile * 16 + mbase + r;
        s[row * KC + m] = c0[r] + bp[m];
        if (c1v) s[row * KC + 16 + m] = c1[r] + bp[16 + m];
    }
}

// --------------------------------------------------------------- row softmax K=20
__global__ void softmax_kernel(const float* __restrict__ s, float* __restrict__ ssm) {
    int i = blockIdx.x * blockDim.x + threadIdx.x;
    if (i >= BN) return;
    const float* r = s + i * KC;
    float mx = r[0];
#pragma unroll
    for (int k = 1; k < KC; ++k) mx = fmaxf(mx, r[k]);
    float sum = 0.f, e[KC];
#pragma unroll
    for (int k = 0; k < KC; ++k) { e[k] = expf(r[k] - mx); sum += e[k]; }
    float inv = 1.0f / sum;
#pragma unroll
    for (int k = 0; k < KC; ++k) ssm[i * KC + k] = e[k] * inv;
}

// ---------------------------------------------------- xp[b,k,h] = sum_n S[n,k]X[n,h]
// one block per batch, 320 threads = (k,h) pairs, 128-node LDS staging
__global__ void xp_kernel(const float* __restrict__ ssm, const float* __restrict__ ne,
                          float* __restrict__ xp) {
    __shared__ float sc[128 * KC];
    __shared__ float xc[128 * HIDC];
    const int b = blockIdx.x, t = threadIdx.x;
    const int k = t / HIDC, h = t % HIDC;
    float acc = 0.f;
    for (int ch = 0; ch < NN / 128; ++ch) {
        int base = b * NN + ch * 128;
        for (int idx = t; idx < 128 * KC; idx += 320) {
            sc[idx] = ssm[(base + idx / KC) * KC + idx % KC];
        }
        for (int idx = t; idx < 128 * HIDC; idx += 320) {
            xc[idx] = ne[(base + idx / HIDC) * HIDC + idx % HIDC];
        }
        __syncthreads();
#pragma unroll 8
        for (int n = 0; n < 128; ++n) acc += sc[n * KC + k] * xc[n * HIDC + h];
        __syncthreads();
    }
    xp[b * KC * HIDC + t] = acc;
}

// --------------------- out_adj[b] += sum_edges s[src] (outer) s[dst]  (20x20/edge)
__global__ void outadj_kernel(const int* __restrict__ src, const int* __restrict__ dst,
                              const float* __restrict__ ssm, float* __restrict__ oadj) {
    __shared__ float si[ECH * KC];
    __shared__ float sj[ECH * KC];
    const int t = threadIdx.x;                       // 512 threads
    const int e0 = blockIdx.x * EPB;
    const int b = e0 >> 15;                          // / (N*DEG)=32768, edges batch-contig
    const bool act = t < KC * KC;
    const int k1 = t / KC, k2 = t % KC;
    float acc = 0.f;
    for (int st = 0; st < EPB / ECH; ++st) {
        int eb = e0 + st * ECH;
        for (int idx = t; idx < ECH * KC; idx += 512) {
            int e = idx / KC, kk = idx % KC;
            si[idx] = ssm[src[eb + e] * KC + kk];
            sj[idx] = ssm[dst[eb + e] * KC + kk];
        }
        __syncthreads();
        if (act) {
#pragma unroll 8
            for (int e = 0; e < ECH; ++e) acc += si[e * KC + k1] * sj[e * KC + k2];
        }
        __syncthreads();
    }
    if (act) atomicAdd(&oadj[b * KC * KC + t], acc);
}

// ------------------------------------- per-batch tail: normalize, conv, heads
__global__ void finalize_kernel(const float* __restrict__ xp, const float* __restrict__ oadj,
                                const float* __restrict__ Wrel, const float* __restrict__ brel,
                                const float* __restrict__ Wroot,
                                const float* __restrict__ Wl1, const float* __restrict__ bl1,
                                const float* __restrict__ Wl2, const float* __restrict__ bl2,
                                float* __restrict__ out, float* __restrict__ gemb) {
    __shared__ float adjp[KC * KC], xps[KC * HIDC], ys[KC * HIDC],
                     xcs[KC * HIDC], hh[KC * HIDC], dd[KC];
    const int b = blockIdx.x, t = threadIdx.x;       // 640 threads
    if (t < KC * KC) {
        float v = oadj[b * KC * KC + t];
        if (t / KC == t % KC) v = 0.0f;              // zero diagonal
        adjp[t] = v;
    }
    if (t < KC * HIDC) xps[t] = xp[b * KC * HIDC + t];
    __syncthreads();
    if (t < KC) {
        float sum = 0.f;
#pragma unroll
        for (int k2 = 0; k2 < KC; ++k2) sum += adjp[t * KC + k2];
        dd[t] = sqrtf(sum) + 1e-15f;
    }
    __syncthreads();
    if (t < KC * KC) adjp[t] = adjp[t] / (dd[t / KC] * dd[t % KC]);
    if (t < KC * HIDC) {                             // y = xp @ Wrel
        int k = t / HIDC, h = t % HIDC;
        float y = 0.f;
#pragma unroll
        for (int h2 = 0; h2 < HIDC; ++h2) y += xps[k * HIDC + h2] * Wrel[h2 * HIDC + h];
        ys[t] = y;
    }
    __syncthreads();
    if (t < KC * HIDC) {                             // xc = adjp@y + brel + xp@Wroot
        int k = t / HIDC, h = t % HIDC;
        float a = 0.f;
#pragma unroll
        for (int k2 = 0; k2 < KC; ++k2) a += adjp[k * KC + k2] * ys[k2 * HIDC + h];
        float r = 0.f;
#pragma unroll
        for (int h2 = 0; h2 < HIDC; ++h2) r += xps[k * HIDC + h2] * Wroot[h2 * HIDC + h];
        xcs[t] = a + brel[h] + r;
    }
    __syncthreads();
    if (t < HIDC) {                                  // g_emb = mean_k xc
        float g = 0.f;
#pragma unroll
        for (int k = 0; k < KC; ++k) g += xcs[k * HIDC + t];
        gemb[b * HIDC + t] = g * (1.0f / KC);
    }
    if (t < KC * HIDC) {                             // h = relu(xc@Wl1 + bl1)
        int k = t / HIDC, h = t % HIDC;
        float v = bl1[h];
#pragma unroll
        for (int h2 = 0; h2 < HIDC; ++h2) v += xcs[k * HIDC + h2] * Wl1[h2 * HIDC + h];
        hh[t] = fmaxf(v, 0.0f);
    }
    __syncthreads();
    {                                                // out = h@Wl2 + bl2 (640 = 20*32)
        int k = t / OUTC, o = t % OUTC;
        float v = bl2[o];
#pragma unroll
        for (int h = 0; h < HIDC; ++h) v += hh[k * HIDC + h] * Wl2[h * OUTC + o];
        out[b * KC * OUTC + k * OUTC + o] = v;
    }
}

extern "C" void kernel_launch(void* const* d_in, const int* in_sizes, int n_in,
                              void* d_out, int out_size, void* d_ws, size_t ws_size,
                              hipStream_t stream) {
    const float* x     = (const float*)d_in[0];
    const int*   ei    = (const int*)d_in[1];
    const float* W1    = (const float*)d_in[3];
    const float* b1    = (const float*)d_in[4];
    const float* Wp    = (const float*)d_in[5];
    const float* bp    = (const float*)d_in[6];
    const float* Wrel  = (const float*)d_in[7];
    const float* brel  = (const float*)d_in[8];
    const float* Wroot = (const float*)d_in[9];
    const float* Wl1   = (const float*)d_in[10];
    const float* bl1   = (const float*)d_in[11];
    const float* Wl2   = (const float*)d_in[12];
    const float* bl2   = (const float*)d_in[13];

    const int E = in_sizes[1] / 2;                 // 1,048,576
    const int* src = ei;
    const int* dst = ei + E;

    // output layout: out | g_emb | node_emb | s
    float* out_p  = (float*)d_out;                           // 32*20*32
    float* gemb_p = out_p + BB * KC * OUTC;                  // 32*16
    float* nemb_p = gemb_p + BB * HIDC;                      // 65536*16
    float* s_p    = nemb_p + BN * HIDC;                      // 65536*20

    // workspace layout (floats)
    float* W    = (float*)d_ws;
    float* h0   = W;                       // 65536*16
    float* agg  = h0   + BN * HIDC;        // 65536*16
    float* deg  = agg  + BN * HIDC;        // 65536
    float* dinv = deg  + BN;               // 65536
    float* ssm  = dinv + BN;               // 65536*20
    float* xp   = ssm  + BN * KC;          // 32*20*16
    float* oadj = xp   + BB * KC * HIDC;   // 32*20*20

    fill_zero_kernel<<<(BN + 255) / 256, 256, 0, stream>>>(deg, BN);
    fill_zero_kernel<<<(BB * KC * KC + 255) / 256, 256, 0, stream>>>(oadj, BB * KC * KC);

    gemm1_wmma_kernel<<<BN / 128, 256, 0, stream>>>(x, W1, h0);
    deg_kernel<<<(E + 255) / 256, 256, 0, stream>>>(dst, deg, E);
    dinv_self_kernel<<<BN / 256, 256, 0, stream>>>(deg, dinv, h0, agg);
    scatter_kernel<<<(E * HIDC) / 256, 256, 0, stream>>>(src, dst, dinv, h0, agg, E);
    bias_relu_kernel<<<(BN * HIDC) / 256, 256, 0, stream>>>(agg, b1, nemb_p);

    sgemm_wmma_kernel<<<BN / 128, 256, 0, stream>>>(nemb_p, Wp, bp, s_p);
    softmax_kernel<<<BN / 256, 256, 0, stream>>>(s_p, ssm);

    xp_kernel<<<BB, 320, 0, stream>>>(ssm, nemb_p, xp);
    outadj_kernel<<<E / EPB, 512, 0, stream>>>(src, dst, ssm, oadj);

    finalize_kernel<<<BB, KC * OUTC, 0, stream>>>(xp, oadj, Wrel, brel, Wroot,
                                                  Wl1, bl1, Wl2, bl2, out_p, gemb_p);
}